// ReconstructionHead_50328426774770
// MI455X (gfx1250) — compile-verified
//
#include <hip/hip_runtime.h>

typedef __attribute__((ext_vector_type(2))) float v2f;
typedef __attribute__((ext_vector_type(4))) float f4;
typedef __attribute__((ext_vector_type(8))) float v8f;
typedef __attribute__((ext_vector_type(4))) unsigned int u32x4;
typedef __attribute__((ext_vector_type(8))) int i32x8;
typedef __attribute__((ext_vector_type(4))) int i32x4;

// Device-pass-only capability checks (host clang doesn't report aux-target builtins).
#if defined(__HIP_DEVICE_COMPILE__)
#if !__has_builtin(__builtin_amdgcn_wmma_f32_16x16x4_f32)
#error "missing __builtin_amdgcn_wmma_f32_16x16x4_f32 for gfx1250"
#endif
#if __has_builtin(__builtin_amdgcn_tensor_load_to_lds) && __has_builtin(__builtin_amdgcn_s_wait_tensorcnt)
#define USE_TDM 1
#endif
#endif

// Static problem configuration (matches reference)
#define Bsz    16384
#define Fn     64
#define Dd     32
#define Hh     64
#define CMAXc  32
#define TOTALc 512

#define ROWS_PER_BLOCK 128   // 8 waves x 16 rows
#define WAVES 8

// LDS row strides in floats (padded; TDM pad_interval/pad_amount reproduce these)
#define IN_STRIDE 36   // 32 + 4
#define W1_STRIDE 80   // 64 + 16
#define W2_STRIDE 40   // 32 + 8
#define H_STRIDE  66

#ifdef USE_TDM
// Issue one 2D TDM tile load: global (row-major, stride0 elements/row) -> LDS,
// with hardware row padding. data_size fixed to 4 bytes (f32).
__device__ __forceinline__ void tdm_load_2d_f32(const void* gptr, unsigned lds_off,
                                                unsigned dim0, unsigned dim1,
                                                unsigned tile0, unsigned tile1,
                                                unsigned stride0,
                                                unsigned pad_interval, unsigned pad_amount) {
    unsigned long long ga = (unsigned long long)(uintptr_t)gptr;
    u32x4 g0;
    g0[0] = 1u;                                   // count=1, user descriptor
    g0[1] = lds_off;                              // lds_addr (bytes)
    g0[2] = (unsigned)(ga & 0xFFFFFFFFu);         // global_addr[31:0]
    g0[3] = (unsigned)((ga >> 32) & 0x1FFFFFFu)   // global_addr[56:32]
          | (2u << 30);                           // type=2 ("image")
    i32x8 g1;
    g1[0] = (int)((2u << 16)                      // data_size = 4B
                | (1u << 20)                      // pad_enable
                | (pad_interval << 22)
                | (pad_amount << 25));
    g1[1] = (int)((dim0 & 0xFFFFu) << 16);                                  // tensor_dim0 lo16
    g1[2] = (int)(((dim0 >> 16) & 0xFFFFu) | ((dim1 & 0xFFFFu) << 16));     // dim0 hi / dim1 lo
    g1[3] = (int)(((dim1 >> 16) & 0xFFFFu) | ((tile0 & 0xFFFFu) << 16));    // dim1 hi / tile_dim0
    g1[4] = (int)(tile1 & 0xFFFFu);               // tile_dim1 (tile_dim2 = 0 -> 2D)
    g1[5] = (int)stride0;                         // tensor_dim0_stride lo32
    g1[6] = 0;                                    // stride0 hi / dim1_stride (unused)
    g1[7] = 0;
    i32x4 z4 = {0, 0, 0, 0};
    i32x8 z8 = {0, 0, 0, 0, 0, 0, 0, 0};
    // clang-23 / therock-10.0 arity: (g0, g1, g2, g3, extra, cpol)
    __builtin_amdgcn_tensor_load_to_lds(g0, g1, z4, z4, z8, 0);
}
#endif

__global__ __launch_bounds__(256)
void recon_head_wmma_f32(const float* __restrict__ inp,   // [B, F, D]
                         const float* __restrict__ W1,    // [F, D, H]
                         const float* __restrict__ b1,    // [F, H]
                         const float* __restrict__ W2,    // [F, H, CMAX]
                         const float* __restrict__ b2,    // [F, CMAX]
                         float* __restrict__ out) {       // [B, TOTAL]
    __shared__ float lds_in[ROWS_PER_BLOCK * IN_STRIDE];  // 18,432 B
    __shared__ float lds_w1[Dd * W1_STRIDE];              // 10,240 B
    __shared__ float lds_w2[Hh * W2_STRIDE];              // 10,240 B
    __shared__ float lds_h [WAVES * 16 * H_STRIDE];       // 33,792 B
    __shared__ float lds_b1[Hh];
    __shared__ float lds_b2[CMAXc];

    const int tid  = threadIdx.x;
    const int wave = tid >> 5;
    const int lane = tid & 31;
    const int f    = blockIdx.y;
    const int rowBase = blockIdx.x * ROWS_PER_BLOCK;

    // Per-feature output card / column offset (closed form of FLAT_IDX)
    int card, off;
    if (f < 32) { int r = f & 3; card = 4 << r; off = (f >> 2) * 60 + ((4 << r) - 4); }
    else        { card = 1;      off = 480 + (f - 32); }

    const float* inpF = inp + (size_t)f * Dd;
    const float* w1F  = W1  + (size_t)f * (Dd * Hh);
    const float* w2F  = W2  + (size_t)f * (Hh * CMAXc);

#ifdef USE_TDM
    // ---- async Tensor Data Mover staging: 3 concurrent 2D tile DMAs with HW padding ----
    if (wave == 0) {
        // input tile: 128 rows x 32 f32, global row stride 2048 el, LDS stride 36 (pad 4 dw / 32 dw)
        tdm_load_2d_f32(inpF + (size_t)rowBase * (Fn * Dd),
                        (unsigned)(uintptr_t)&lds_in[0],
                        Dd, ROWS_PER_BLOCK, Dd, ROWS_PER_BLOCK, Fn * Dd, 4, 3);
        __builtin_amdgcn_s_wait_tensorcnt(0);
    } else if (wave == 1) {
        // W1: 32 rows x 64 f32 contiguous, LDS stride 80 (pad 16 dw / 64 dw)
        tdm_load_2d_f32(w1F, (unsigned)(uintptr_t)&lds_w1[0],
                        Hh, Dd, Hh, Dd, Hh, 5, 15);
        __builtin_amdgcn_s_wait_tensorcnt(0);
    } else if (wave == 2) {
        // W2: 64 rows x 32 f32 contiguous, LDS stride 40 (pad 8 dw / 32 dw)
        tdm_load_2d_f32(w2F, (unsigned)(uintptr_t)&lds_w2[0],
                        CMAXc, Hh, CMAXc, Hh, CMAXc, 4, 7);
        __builtin_amdgcn_s_wait_tensorcnt(0);
    }
#else
    // ---- fallback: vector-load staging ----
    #pragma unroll
    for (int i = 0; i < 4; ++i) {
        int s = tid + i * 256;
        int row = s >> 3, part = s & 7;
        f4 v = *(const f4*)(inpF + (size_t)(rowBase + row) * (Fn * Dd) + part * 4);
        *(f4*)(&lds_in[row * IN_STRIDE + part * 4]) = v;
    }
    #pragma unroll
    for (int i = 0; i < 2; ++i) {
        int s = tid + i * 256;
        int d = s >> 4, hseg = s & 15;
        f4 v = *(const f4*)(w1F + d * Hh + hseg * 4);
        *(f4*)(&lds_w1[d * W1_STRIDE + hseg * 4]) = v;
    }
    #pragma unroll
    for (int i = 0; i < 2; ++i) {
        int s = tid + i * 256;
        int h = s >> 3, cseg = s & 7;
        f4 v = *(const f4*)(w2F + h * CMAXc + cseg * 4);
        *(f4*)(&lds_w2[h * W2_STRIDE + cseg * 4]) = v;
    }
#endif
    // biases staged by waves 3..5 (do not collide with TDM-issuing waves)
    if (tid >= 96 && tid < 96 + Hh)                  lds_b1[tid - 96]         = b1[(size_t)f * Hh + (tid - 96)];
    else if (tid >= 96 + Hh && tid < 96 + Hh + CMAXc) lds_b2[tid - 96 - Hh]   = b2[(size_t)f * CMAXc + (tid - 96 - Hh)];

    __syncthreads();

    const int mrow  = lane & 15;   // M (A-layout) or N (B/C-layout) index
    const int khalf = lane >> 4;   // which half of the K quad this lane holds

    // ======== GEMM1: [16x32] x [32x64] -> 16x64 (4 N-tiles, 8 K-steps of 4) ========
    v8f acc1[4] = {};
    for (int ks = 0; ks < 8; ++ks) {
        const int k0 = ks * 4 + khalf * 2;
        v2f a = *(const v2f*)(&lds_in[mrow * IN_STRIDE + k0]);
        #pragma unroll
        for (int nt = 0; nt < 4; ++nt) {
            v2f b;
            b.x = lds_w1[(k0 + 0) * W1_STRIDE + nt * 16 + mrow];
            b.y = lds_w1[(k0 + 1) * W1_STRIDE + nt * 16 + mrow];
            acc1[nt] = __builtin_amdgcn_wmma_f32_16x16x4_f32(
                false, a, false, b, (short)0, acc1[nt], false, false);
        }
    }

    // ---- bias + ReLU, write h tile to per-wave LDS region (C-layout -> memory) ----
    float* hbase = &lds_h[wave * 16 * H_STRIDE];
    #pragma unroll
    for (int nt = 0; nt < 4; ++nt) {
        const int n = nt * 16 + mrow;
        const float bias = lds_b1[n];
        #pragma unroll
        for (int r = 0; r < 8; ++r) {
            const int m = r + khalf * 8;
            float v = acc1[nt][r] + bias;
            hbase[m * H_STRIDE + n] = v > 0.f ? v : 0.f;
        }
    }
    __syncthreads();

    // ======== GEMM2: [16x64] x [64x32] -> 16x32 (<=2 N-tiles, 16 K-steps of 4) ========
    const int ntmax = (card > 16) ? 2 : 1;
    v8f acc2[2] = {};
    for (int ks = 0; ks < 16; ++ks) {
        const int k0 = ks * 4 + khalf * 2;
        v2f a = *(const v2f*)(&hbase[mrow * H_STRIDE + k0]);
        for (int nt = 0; nt < ntmax; ++nt) {
            v2f b;
            b.x = lds_w2[(k0 + 0) * W2_STRIDE + nt * 16 + mrow];
            b.y = lds_w2[(k0 + 1) * W2_STRIDE + nt * 16 + mrow];
            acc2[nt] = __builtin_amdgcn_wmma_f32_16x16x4_f32(
                false, a, false, b, (short)0, acc2[nt], false, false);
        }
    }

    // ---- bias + scatter only the valid (unpadded) columns: out[:, off : off+card] ----
    for (int nt = 0; nt < ntmax; ++nt) {
        const int n = nt * 16 + mrow;
        if (n < card) {
            const float bias = lds_b2[n];
            #pragma unroll
            for (int r = 0; r < 8; ++r) {
                const int m   = r + khalf * 8;
                const int row = rowBase + wave * 16 + m;
                out[(size_t)row * TOTALc + off + n] = acc2[nt][r] + bias;
            }
        }
    }
}

extern "C" void kernel_launch(void* const* d_in, const int* in_sizes, int n_in,
                              void* d_out, int out_size, void* d_ws, size_t ws_size,
                              hipStream_t stream) {
    (void)in_sizes; (void)n_in; (void)out_size; (void)d_ws; (void)ws_size;
    const float* inp = (const float*)d_in[0];
    const float* W1  = (const float*)d_in[1];
    const float* b1  = (const float*)d_in[2];
    const float* W2  = (const float*)d_in[3];
    const float* b2  = (const float*)d_in[4];
    float* out = (float*)d_out;

    dim3 grid(Bsz / ROWS_PER_BLOCK, Fn);   // (128, 64)
    recon_head_wmma_f32<<<grid, 256, 0, stream>>>(inp, W1, b1, W2, b2, out);
}